// Attention_44487271252662
// MI455X (gfx1250) — compile-verified
//
#include <hip/hip_runtime.h>

typedef float v2f __attribute__((ext_vector_type(2)));
typedef float v4f __attribute__((ext_vector_type(4)));
typedef float v8f __attribute__((ext_vector_type(8)));

#define FEA 1024
#define SEQ 2048
#define NB  32

// -----------------------------------------------------------------------------
// Kernel 1: dec_fea = s_t @ dec_W^T + dec_b   (32x1024 = 32x1024 * 1024x1024^T)
// One wave (32 threads) per 16x16 output tile, V_WMMA_F32_16X16X4_F32 over K.
// A layout (16x4 f32): vgpr v, lane-group g (lane>=16) holds K = 2g + v, M = lane%16
// B layout (4x16 f32): symmetric, N = lane%16. C/D: vgpr v -> M = v + 8g, N = lane%16.
// -----------------------------------------------------------------------------
__global__ __launch_bounds__(32)
void dec_fc_wmma(const float* __restrict__ s_t, const float* __restrict__ dec_W,
                 const float* __restrict__ dec_b, float* __restrict__ dec_fea) {
  const int lane   = threadIdx.x;
  const int n_base = blockIdx.x * 16;   // 64 tiles over F
  const int m_base = blockIdx.y * 16;   // 2 tiles over B
  const int lg   = lane >> 4;           // lane group 0/1
  const int lm   = lane & 15;
  const int koff = lg * 2;              // this lane's K pair offset within a k4 step

  const float* arow = s_t  + (size_t)(m_base + lm) * FEA;   // A row m (k-major)
  const float* brow = dec_W + (size_t)(n_base + lm) * FEA;  // B col n == dec_W row n

  v8f acc = {0.f, 0.f, 0.f, 0.f, 0.f, 0.f, 0.f, 0.f};
#pragma unroll 4
  for (int kk = 0; kk < FEA; kk += 4) {
    v2f a = *(const v2f*)(arow + kk + koff);
    v2f b = *(const v2f*)(brow + kk + koff);
    acc = __builtin_amdgcn_wmma_f32_16x16x4_f32(
        /*neg_a=*/false, a, /*neg_b=*/false, b,
        /*c_mod=*/(short)0, acc, /*reuse_a=*/false, /*reuse_b=*/false);
  }

  const int   col  = n_base + lm;
  const float bias = dec_b[col];
#pragma unroll
  for (int v = 0; v < 8; ++v) {
    const int row = m_base + v + 8 * lg;
    dec_fea[(size_t)row * FEA + col] = acc[v] + bias;
  }
}

// -----------------------------------------------------------------------------
// Kernel 2: scores[b,s] = sum_f tanh(enc_fea + dec_fea[b] + cov[b,s]*cov_w) * att_w
// One wave32 per row; streams 256 MB of enc_fea once -> non-temporal loads so the
// single-use stream does not evict the reused small tensors from L2.
// tanh via hardware exp: tanh(x) = 1 - 2/(exp(2x)+1)  (saturates correctly at +/-1)
// -----------------------------------------------------------------------------
__device__ __forceinline__ float fast_tanh(float x) {
  const float e = __expf(2.0f * x);
  return 1.0f - 2.0f / (e + 1.0f);
}

__global__ __launch_bounds__(256)
void scores_kernel(const float* __restrict__ enc_fea, const float* __restrict__ dec_fea,
                   const float* __restrict__ coverage, const float* __restrict__ cov_w,
                   const float* __restrict__ att_w, float* __restrict__ scores) {
  const int wave = threadIdx.x >> 5;
  const int lane = threadIdx.x & 31;
  const int r    = blockIdx.x * 8 + wave;    // row in [0, B*S)
  const int b    = r >> 11;                  // S = 2048
  const float cov = coverage[r];

  const v4f* ef = (const v4f*)(enc_fea + (size_t)r * FEA);
  const v4f* df = (const v4f*)(dec_fea + (size_t)b * FEA);
  const v4f* cw = (const v4f*)cov_w;
  const v4f* aw = (const v4f*)att_w;

  float part = 0.f;
#pragma unroll
  for (int jj = 0; jj < FEA / 4 / 32; ++jj) {   // 8 iterations, 32 floats/lane
    const int j = lane + jj * 32;
    const v4f e4 = __builtin_nontemporal_load(ef + j);  // single-use stream: TH=NT
    const v4f d4 = df[j];                               // reused: keep cached
    const v4f c4 = cw[j];
    const v4f a4 = aw[j];
    const v4f x  = e4 + d4 + cov * c4;
    part += fast_tanh(x.x) * a4.x;
    part += fast_tanh(x.y) * a4.y;
    part += fast_tanh(x.z) * a4.z;
    part += fast_tanh(x.w) * a4.w;
  }
  for (int off = 16; off > 0; off >>= 1)
    part += __shfl_down(part, off, 32);
  if (lane == 0) scores[r] = part;
}

// -----------------------------------------------------------------------------
// Kernel 3: attn = exp(scores - max)*mask / sum(...); cov_next = coverage + attn
// (softmax -> mask -> renormalize folds into one masked softmax)
// One block per batch row b; 256 threads x 8 elements.
// -----------------------------------------------------------------------------
__global__ __launch_bounds__(256)
void softmax_kernel(const float* __restrict__ scores, const float* __restrict__ mask,
                    const float* __restrict__ coverage,
                    float* __restrict__ attn_out, float* __restrict__ cov_out) {
  __shared__ float red[256];
  const int b   = blockIdx.x;
  const int tid = threadIdx.x;
  const float* sc = scores + (size_t)b * SEQ;

  float vals[8];
  float mx = -3.4e38f;
#pragma unroll
  for (int i = 0; i < 8; ++i) {
    vals[i] = sc[tid + i * 256];
    mx = fmaxf(mx, vals[i]);
  }
  red[tid] = mx; __syncthreads();
  for (int s = 128; s > 0; s >>= 1) {
    if (tid < s) red[tid] = fmaxf(red[tid], red[tid + s]);
    __syncthreads();
  }
  mx = red[0];
  __syncthreads();

  float e[8];
  float sum = 0.f;
#pragma unroll
  for (int i = 0; i < 8; ++i) {
    const int s = tid + i * 256;
    e[i] = __expf(vals[i] - mx) * mask[(size_t)b * SEQ + s];
    sum += e[i];
  }
  red[tid] = sum; __syncthreads();
  for (int s = 128; s > 0; s >>= 1) {
    if (tid < s) red[tid] += red[tid + s];
    __syncthreads();
  }
  const float inv = 1.0f / red[0];

#pragma unroll
  for (int i = 0; i < 8; ++i) {
    const int s = tid + i * 256;
    const float a = e[i] * inv;
    attn_out[(size_t)b * SEQ + s] = a;
    cov_out[(size_t)b * SEQ + s]  = coverage[(size_t)b * SEQ + s] + a;
  }
}

// -----------------------------------------------------------------------------
// Kernel 4: c_t partials. Grid (s_chunk=8, b=32); block 256 threads.
// Each lane owns one float4 column group (block covers all F=1024); attn chunk
// cached in LDS; enc_out streamed once with b128 non-temporal loads (256 MB).
// -----------------------------------------------------------------------------
__global__ __launch_bounds__(256)
void ctx_partial_kernel(const float* __restrict__ attn, const float* __restrict__ enc_out,
                        float* __restrict__ partial) {
  __shared__ float a_sh[256];
  const int tid = threadIdx.x;
  const int sch = blockIdx.x;   // 0..7
  const int b   = blockIdx.y;   // 0..31
  const int s0  = sch * 256;

  a_sh[tid] = attn[(size_t)b * SEQ + s0 + tid];
  __syncthreads();

  const v4f* base = (const v4f*)(enc_out + ((size_t)b * SEQ + s0) * FEA) + tid;
  v4f acc = {0.f, 0.f, 0.f, 0.f};
#pragma unroll 4
  for (int s = 0; s < 256; ++s) {
    const v4f e = __builtin_nontemporal_load(base + (size_t)s * (FEA / 4));
    acc += a_sh[s] * e;
  }

  ((v4f*)(partial + ((size_t)(b * 8 + sch)) * FEA))[tid] = acc;
}

// -----------------------------------------------------------------------------
// Kernel 5: reduce the 8 s-chunk partials -> c_t (32x1024)
// -----------------------------------------------------------------------------
__global__ __launch_bounds__(256)
void ctx_reduce_kernel(const float* __restrict__ partial, float* __restrict__ c_t) {
  const int i = blockIdx.x * 256 + threadIdx.x;   // 0..32767
  const int b = i >> 10, f = i & 1023;
  float acc = 0.f;
#pragma unroll
  for (int sch = 0; sch < 8; ++sch)
    acc += partial[((size_t)(b * 8 + sch)) * FEA + f];
  c_t[i] = acc;
}

// -----------------------------------------------------------------------------
extern "C" void kernel_launch(void* const* d_in, const int* in_sizes, int n_in,
                              void* d_out, int out_size, void* d_ws, size_t ws_size,
                              hipStream_t stream) {
  const float* s_t      = (const float*)d_in[0];  // (32,1024)
  const float* enc_out  = (const float*)d_in[1];  // (32,2048,1024)
  const float* enc_fea  = (const float*)d_in[2];  // (65536,1024)
  const float* mask     = (const float*)d_in[3];  // (32,2048)
  const float* coverage = (const float*)d_in[4];  // (32,2048)
  const float* dec_W    = (const float*)d_in[5];  // (1024,1024)
  const float* dec_b    = (const float*)d_in[6];  // (1024,)
  const float* att_w    = (const float*)d_in[7];  // (1024,)
  const float* cov_w    = (const float*)d_in[8];  // (1024,)

  float* out   = (float*)d_out;
  float* c_t   = out;                 // 32*1024  = 32768
  float* attn  = out + 32768;         // 32*2048  = 65536
  float* covn  = out + 32768 + 65536; // 32*2048  = 65536

  float* ws      = (float*)d_ws;
  float* dec_fea = ws;                 // 32768 floats
  float* scores  = ws + 32768;         // 65536 floats
  float* partial = ws + 32768 + 65536; // 32*8*1024 = 262144 floats (total 1.5 MB)

  dec_fc_wmma       <<<dim3(64, 2),   32, 0, stream>>>(s_t, dec_W, dec_b, dec_fea);
  scores_kernel     <<<8192,         256, 0, stream>>>(enc_fea, dec_fea, coverage,
                                                       cov_w, att_w, scores);
  softmax_kernel    <<<NB,           256, 0, stream>>>(scores, mask, coverage, attn, covn);
  ctx_partial_kernel<<<dim3(8, 32),  256, 0, stream>>>(attn, enc_out, partial);
  ctx_reduce_kernel <<<128,          256, 0, stream>>>(partial, c_t);
}